// Encoder_21921513079181
// MI455X (gfx1250) — compile-verified
//
#include <hip/hip_runtime.h>
#include <math.h>

// ---------------------------------------------------------------------------
// Transformer encoder (2 blocks), fp32, for MI455X (gfx1250).
// All matmuls run on V_WMMA_F32_16X16X4_F32 (full-precision WMMA).
// Attention is fused (flash-style, online softmax) so the 512MB/block score
// tensor never touches HBM. GEMMs are register-blocked 16x64 per wave (one
// A-frag feeds 4 WMMAs) and K is a template constant so all operand
// addresses fold into load immediates; each k-slab is load-phase then
// WMMA-phase so the scheduler can batch loads and stagger s_wait_loadcnt.
// ---------------------------------------------------------------------------

typedef float v2f __attribute__((ext_vector_type(2)));
typedef float v8f __attribute__((ext_vector_type(8)));

constexpr int kB   = 4;
constexpr int kS   = 2048;
constexpr int kD   = 256;
constexpr int kH   = 8;
constexpr int kDK  = 32;      // kD / kH
constexpr int kDFF = 1024;
constexpr int kM   = kB * kS; // 8192 rows

__device__ __forceinline__ v8f wmma_f32(v2f a, v2f b, v8f c) {
  // D = A(16x4 f32) * B(4x16 f32) + C(16x16 f32)
  return __builtin_amdgcn_wmma_f32_16x16x4_f32(
      /*neg_a=*/false, a, /*neg_b=*/false, b,
      /*c_mod=*/(short)0, c, /*reuse_a=*/false, /*reuse_b=*/false);
}

// ---------------------------------------------------------------------------
// Y[m][n] = relu( sum_k X[m][k] * W[n][k] )   (torch F.linear weight layout)
// One wave per 16(M) x 64(N) output tile: 4 accumulators share one A-frag.
// K compile-time (256 or 1024), N multiple of 64.
// A-frag lane map: m = lane&15, k-pair = 2*(lane>>4)   (ISA 16x4 f32 layout)
// B-frag lane map: n = lane&15, k-pair = 2*(lane>>4)
// C/D lane map: row = r + 8*(lane>>4), col = lane&15, r = VGPR index 0..7
// ---------------------------------------------------------------------------
template <int K>
__global__ void gemm_relu_wmma(const float* __restrict__ X,
                               const float* __restrict__ W,
                               float* __restrict__ Y,
                               int N) {
  const int lane = threadIdx.x & 31;
  const int wave = threadIdx.x >> 5;
  const int tile = blockIdx.x * (blockDim.x >> 5) + wave;
  const int NT   = N >> 6;                 // 64-wide N tiles
  const int mt   = tile / NT;
  const int nt   = tile - mt * NT;
  const int mrow = lane & 15;
  const int t2   = (lane >> 4) << 1;

  const float* xrow = X + (size_t)(mt * 16 + mrow) * K + t2;
  const float* wrow = W + (size_t)(nt * 64 + mrow) * K + t2;

  v8f acc0 = {}, acc1 = {}, acc2 = {}, acc3 = {};

  for (int k = 0; k < K; k += 16) {
    v2f a[4], b0[4], b1[4], b2[4], b3[4];
    // ---- load phase: 20 b64 loads in flight (constant immediate offsets)
#pragma unroll
    for (int u = 0; u < 4; ++u) {
      a[u]  = *(const v2f*)(xrow + k + 4 * u);
      b0[u] = *(const v2f*)(wrow + 0 * 16 * K + k + 4 * u);
      b1[u] = *(const v2f*)(wrow + 1 * 16 * K + k + 4 * u);
      b2[u] = *(const v2f*)(wrow + 2 * 16 * K + k + 4 * u);
      b3[u] = *(const v2f*)(wrow + 3 * 16 * K + k + 4 * u);
    }
    // ---- compute phase: 16 WMMAs
#pragma unroll
    for (int u = 0; u < 4; ++u) {
      acc0 = wmma_f32(a[u], b0[u], acc0);
      acc1 = wmma_f32(a[u], b1[u], acc1);
      acc2 = wmma_f32(a[u], b2[u], acc2);
      acc3 = wmma_f32(a[u], b3[u], acc3);
    }
  }

  const int row0 = mt * 16 + ((lane >> 4) << 3);
  const int col  = nt * 64 + mrow;
#pragma unroll
  for (int r = 0; r < 8; ++r) {
    float* yrow = Y + (size_t)(row0 + r) * N + col;
    yrow[0]  = fmaxf(acc0[r], 0.0f);
    yrow[16] = fmaxf(acc1[r], 0.0f);
    yrow[32] = fmaxf(acc2[r], 0.0f);
    yrow[48] = fmaxf(acc3[r], 0.0f);
  }
}

// ---------------------------------------------------------------------------
// Fused attention: O = softmax(QK^T/sqrt(DK)) V  per (b,h), online softmax.
// One wave per (b, h, 16-row q tile). Q/K/V/O are [B,S,D] with heads
// interleaved in the last dim (dk slot = h*kDK + dk).
// ---------------------------------------------------------------------------
__global__ void flash_attn_wmma(const float* __restrict__ Q,
                                const float* __restrict__ Km,
                                const float* __restrict__ V,
                                float* __restrict__ O) {
  const int lane = threadIdx.x & 31;
  const int mrow = lane & 15;
  const int t2   = (lane >> 4) << 1;
  const int QT   = kS / 16;

  const int idx = blockIdx.x;
  const int qt  = idx % QT;
  const int bh  = idx / QT;
  const int h   = bh % kH;
  const int b   = bh / kH;

  const float scale = 0.17677669529663687f; // 1/sqrt(32)

  // Preload Q fragments (A-layout), pre-scaled.
  v2f aq[8];
  const float* qrow = Q + (size_t)(b * kS + qt * 16 + mrow) * kD + h * kDK + t2;
#pragma unroll
  for (int st = 0; st < 8; ++st) {
    v2f a = *(const v2f*)(qrow + st * 4);
    aq[st] = a * scale;
  }

  v8f o0 = {}, o1 = {};
  float mst[8], lst[8];
#pragma unroll
  for (int r = 0; r < 8; ++r) { mst[r] = -3.0e38f; lst[r] = 0.0f; }

  __shared__ float P[16 * 16]; // probability tile (C-layout -> A-layout relay)

  for (int kt = 0; kt < QT; ++kt) {
    const int key0 = kt * 16;

    // ---- S-tile = (Q*scale) @ K^T : load all 8 B-frags, then 8 WMMAs
    const float* krow = Km + (size_t)(b * kS + key0 + mrow) * kD + h * kDK + t2;
    v2f bk[8];
#pragma unroll
    for (int st = 0; st < 8; ++st) bk[st] = *(const v2f*)(krow + st * 4);
    v8f s = {};
#pragma unroll
    for (int st = 0; st < 8; ++st) s = wmma_f32(aq[st], bk[st], s);

    // Online softmax update; row r+8*(lane>>4) lives across one 16-lane half.
#pragma unroll
    for (int r = 0; r < 8; ++r) {
      float v  = s[r];
      float mx = v;
      mx = fmaxf(mx, __shfl_xor(mx, 1, 16));
      mx = fmaxf(mx, __shfl_xor(mx, 2, 16));
      mx = fmaxf(mx, __shfl_xor(mx, 4, 16));
      mx = fmaxf(mx, __shfl_xor(mx, 8, 16));
      float mnew  = fmaxf(mst[r], mx);
      float alpha = __expf(mst[r] - mnew);
      float p     = __expf(v - mnew);
      float rs = p;
      rs += __shfl_xor(rs, 1, 16);
      rs += __shfl_xor(rs, 2, 16);
      rs += __shfl_xor(rs, 4, 16);
      rs += __shfl_xor(rs, 8, 16);
      lst[r] = lst[r] * alpha + rs;
      mst[r] = mnew;
      o0[r] *= alpha;
      o1[r] *= alpha;
      P[(r + ((lane >> 4) << 3)) * 16 + mrow] = p; // C-layout scatter to LDS
    }
    __syncthreads(); // single-wave WG: lowers to S_NOP, orders LDS

    // ---- O += P @ V_tile: hoist all LDS + V loads, then 8 WMMAs
    const float* vbase = V + (size_t)(b * kS) * kD + h * kDK;
    v2f ap[4], bv0[4], bv1[4];
#pragma unroll
    for (int st = 0; st < 4; ++st) {
      ap[st] = *(const v2f*)(&P[mrow * 16 + st * 4 + t2]); // A-layout gather
      const float* vb = vbase + (size_t)(key0 + st * 4 + t2) * kD + mrow;
      bv0[st].x = vb[0];        bv0[st].y = vb[kD];
      bv1[st].x = vb[16];       bv1[st].y = vb[kD + 16];
    }
#pragma unroll
    for (int st = 0; st < 4; ++st) {
      o0 = wmma_f32(ap[st], bv0[st], o0);
      o1 = wmma_f32(ap[st], bv1[st], o1);
    }
    __syncthreads();
  }

  const int row0 = (lane >> 4) << 3;
#pragma unroll
  for (int r = 0; r < 8; ++r) {
    float inv = 1.0f / lst[r];
    float* orow = O + (size_t)(b * kS + qt * 16 + row0 + r) * kD + h * kDK;
    orow[mrow]      = o0[r] * inv;
    orow[16 + mrow] = o1[r] * inv;
  }
}

// ---------------------------------------------------------------------------
// y = LayerNorm(a + res) * g + beta  over D=256. One wave per row.
// ---------------------------------------------------------------------------
__global__ void ln_residual(const float* __restrict__ A,
                            const float* __restrict__ R,
                            const float* __restrict__ G,
                            const float* __restrict__ Bt,
                            float* __restrict__ Y) {
  const int lane = threadIdx.x & 31;
  const int row  = blockIdx.x * (blockDim.x >> 5) + (threadIdx.x >> 5);
  const float* a = A + (size_t)row * kD;
  const float* r = R + (size_t)row * kD;

  float v[8];
  float s = 0.0f;
#pragma unroll
  for (int i = 0; i < 8; ++i) {
    int idx = lane + (i << 5);
    v[i] = a[idx] + r[idx];
    s += v[i];
  }
  s += __shfl_xor(s, 1, 32);
  s += __shfl_xor(s, 2, 32);
  s += __shfl_xor(s, 4, 32);
  s += __shfl_xor(s, 8, 32);
  s += __shfl_xor(s, 16, 32);
  const float mu = s * (1.0f / kD);

  float s2 = 0.0f;
#pragma unroll
  for (int i = 0; i < 8; ++i) { float d = v[i] - mu; s2 += d * d; }
  s2 += __shfl_xor(s2, 1, 32);
  s2 += __shfl_xor(s2, 2, 32);
  s2 += __shfl_xor(s2, 4, 32);
  s2 += __shfl_xor(s2, 8, 32);
  s2 += __shfl_xor(s2, 16, 32);
  const float rsd = rsqrtf(s2 * (1.0f / kD) + 1e-5f);

  float* y = Y + (size_t)row * kD;
#pragma unroll
  for (int i = 0; i < 8; ++i) {
    int idx = lane + (i << 5);
    y[idx] = (v[i] - mu) * rsd * G[idx] + Bt[idx];
  }
}

// ---------------------------------------------------------------------------
// Orchestration. Workspace layout (floats): Q,K,V,A,T,X each M*D (8MB),
// F = M*DFF (32MB) -> 80MB total. Every buffer fully written before read.
// ---------------------------------------------------------------------------
extern "C" void kernel_launch(void* const* d_in, const int* in_sizes, int n_in,
                              void* d_out, int out_size, void* d_ws, size_t ws_size,
                              hipStream_t stream) {
  const float* x    = (const float*)d_in[0];
  const float* W11  = (const float*)d_in[1];
  const float* W12  = (const float*)d_in[2];
  const float* W13  = (const float*)d_in[3];
  const float* W14  = (const float*)d_in[4];
  const float* W21  = (const float*)d_in[5];
  const float* W22  = (const float*)d_in[6];
  const float* W23  = (const float*)d_in[7];
  const float* W24  = (const float*)d_in[8];
  const float* Wf11 = (const float*)d_in[9];
  const float* Wf21 = (const float*)d_in[10];
  const float* Wf12 = (const float*)d_in[11];
  const float* Wf22 = (const float*)d_in[12];
  const float* g1 = (const float*)d_in[13];
  const float* b1 = (const float*)d_in[14];
  const float* g2 = (const float*)d_in[15];
  const float* b2 = (const float*)d_in[16];
  const float* g3 = (const float*)d_in[17];
  const float* b3 = (const float*)d_in[18];
  const float* g4 = (const float*)d_in[19];
  const float* b4 = (const float*)d_in[20];

  float* ws = (float*)d_ws;
  const size_t MD = (size_t)kM * kD;
  float* bQ = ws;
  float* bK = bQ + MD;
  float* bV = bK + MD;
  float* bA = bV + MD;
  float* bT = bA + MD;
  float* bX = bT + MD;
  float* bF = bX + MD; // kM * kDFF

  const dim3 blk(256);
  const int gD  = (kM / 16) * (kD / 64)   / 8; // 256 blocks, D-wide GEMMs
  const int gF  = (kM / 16) * (kDFF / 64) / 8; // 1024 blocks, DFF-wide GEMM
  const int gAt = kB * kH * (kS / 16);         // 4096 waves of attention
  const int gLn = kM / 8;                      // 1024 blocks, layernorm

  // ---- Block 1 (x input) ----
  gemm_relu_wmma<kD><<<gD, blk, 0, stream>>>(x, W11, bQ, kD);
  gemm_relu_wmma<kD><<<gD, blk, 0, stream>>>(x, W12, bK, kD);
  gemm_relu_wmma<kD><<<gD, blk, 0, stream>>>(x, W13, bV, kD);
  flash_attn_wmma<<<gAt, dim3(32), 0, stream>>>(bQ, bK, bV, bA);
  gemm_relu_wmma<kD><<<gD, blk, 0, stream>>>(bA, W14, bT, kD);
  ln_residual<<<gLn, blk, 0, stream>>>(bT, x, g1, b1, bX);
  gemm_relu_wmma<kD><<<gF, blk, 0, stream>>>(bX, Wf11, bF, kDFF);
  gemm_relu_wmma<kDFF><<<gD, blk, 0, stream>>>(bF, Wf21, bT, kD);
  ln_residual<<<gLn, blk, 0, stream>>>(bT, bX, g2, b2, bA); // x2 -> bA

  // ---- Block 2 (x := bA) ----
  gemm_relu_wmma<kD><<<gD, blk, 0, stream>>>(bA, W21, bQ, kD);
  gemm_relu_wmma<kD><<<gD, blk, 0, stream>>>(bA, W22, bK, kD);
  gemm_relu_wmma<kD><<<gD, blk, 0, stream>>>(bA, W23, bV, kD);
  flash_attn_wmma<<<gAt, dim3(32), 0, stream>>>(bQ, bK, bV, bX);
  gemm_relu_wmma<kD><<<gD, blk, 0, stream>>>(bX, W24, bT, kD);
  ln_residual<<<gLn, blk, 0, stream>>>(bT, bA, g3, b3, bQ); // x1' -> bQ
  gemm_relu_wmma<kD><<<gF, blk, 0, stream>>>(bQ, Wf12, bF, kDFF);
  gemm_relu_wmma<kDFF><<<gD, blk, 0, stream>>>(bF, Wf22, bT, kD);
  ln_residual<<<gLn, blk, 0, stream>>>(bT, bQ, g4, b4, (float*)d_out);
}